// MoEGate_46420006535177
// MI455X (gfx1250) — compile-verified
//
#include <hip/hip_runtime.h>

// ---------------------------------------------------------------------------
// MoE gate: logits = x @ W^T via V_WMMA_F32_16X16X4_F32, per-row top-8 +
// renormalized softmax (full softmax denominator cancels under renorm).
//   x: (T=16384, H=2048) f32     W: (E=64, H=2048) f32
//   out: [ topk_idx : int32 T*8 ][ topk_weight : f32 T*8 ]
// Global->LDS staging uses CDNA5 async copies (GLOBAL_LOAD_ASYNC_TO_LDS_B128,
// ASYNCcnt) with double buffering.
// ---------------------------------------------------------------------------

typedef __attribute__((ext_vector_type(2))) float v2f;
typedef __attribute__((ext_vector_type(8))) float v8f;

#define T_TOKENS 16384
#define H_DIM    2048
#define E_EXP    64
#define TOPK     8

#define ROWS     64          // tokens per block = 4 waves * 16
#define KC       32          // K-chunk per pipeline stage
#define NCHUNK   (H_DIM / KC)            // 64
#define K4       (KC / 4)                // 8 float4 per row per chunk

// byte strides inside the LDS tiles (padded for conflict-free fragment reads)
#define XROW_B   ((K4 + 1) * 16)         // 144 B per token row
#define WROW_B   ((E_EXP + 1) * 16)      // 1040 B per k4 row
#define XBUF_B   (ROWS * XROW_B)         // 9216 B
#define WBUF_B   (K4 * WROW_B)           // 8320 B

__device__ __forceinline__ void async_ld_b128(unsigned lds_off, const void* gaddr) {
    // GLOBAL_LOAD_ASYNC_TO_LDS_B128: vdst = LDS byte address, vaddr = 64-bit VA
    asm volatile("global_load_async_to_lds_b128 %0, %1, off"
                 :: "v"(lds_off), "v"(gaddr) : "memory");
}

__device__ __forceinline__ void wait_async_le8() {
#if __has_builtin(__builtin_amdgcn_s_wait_asynccnt)
    __builtin_amdgcn_s_wait_asynccnt(8);
#else
    asm volatile("s_wait_asynccnt 0x8" ::: "memory");
#endif
}

__device__ __forceinline__ void wait_async_0() {
#if __has_builtin(__builtin_amdgcn_s_wait_asynccnt)
    __builtin_amdgcn_s_wait_asynccnt(0);
#else
    asm volatile("s_wait_asynccnt 0x0" ::: "memory");
#endif
}

__global__ __launch_bounds__(128)
void moe_gate_topk_kernel(const float* __restrict__ x,
                          const float* __restrict__ w,
                          int*   __restrict__ idx_out,
                          float* __restrict__ w_out)
{
    __shared__ char  xbuf[2][XBUF_B];     // [row][k4] float4, +16B row pad
    __shared__ char  wbuf[2][WBUF_B];     // [k4][e]  float4, +16B row pad
    __shared__ float scores[ROWS][E_EXP];

    const int tid  = threadIdx.x;
    const int lane = tid & 31;
    const int wave = tid >> 5;            // 0..3 -> M-tile
    const int row0 = blockIdx.x * ROWS;

    const int half = lane >> 4;           // 0: (K,K+1) | 1: (K+2,K+3)
    const int l16  = lane & 15;

    // per-thread async-copy coordinates: i = tid + j*128, j in 0..3
    //   k4 = i & 7 (constant per thread), e/r = (tid>>3) + 16*j
    const int k4  = tid & 7;
    const int er0 = tid >> 3;             // 0..15

    const unsigned xlds0 = (unsigned)(uintptr_t)&xbuf[0][0];
    const unsigned xlds1 = (unsigned)(uintptr_t)&xbuf[1][0];
    const unsigned wlds0 = (unsigned)(uintptr_t)&wbuf[0][0];
    const unsigned wlds1 = (unsigned)(uintptr_t)&wbuf[1][0];

    // issue one chunk's async copies (8 x b128 per thread -> 8 per wave ASYNCcnt)
    auto issue_chunk = [&](int kb, int buf) {
        const unsigned xb = buf ? xlds1 : xlds0;
        const unsigned wb = buf ? wlds1 : wlds0;
        #pragma unroll
        for (int j = 0; j < 4; ++j) {
            const int e = er0 + 16 * j;
            async_ld_b128(wb + (unsigned)(k4 * WROW_B + e * 16),
                          w + (size_t)e * H_DIM + kb + k4 * 4);
        }
        #pragma unroll
        for (int j = 0; j < 4; ++j) {
            const int r = er0 + 16 * j;
            async_ld_b128(xb + (unsigned)(r * XROW_B + k4 * 16),
                          x + (size_t)(row0 + r) * H_DIM + kb + k4 * 4);
        }
    };

    v8f acc0 = {}, acc1 = {}, acc2 = {}, acc3 = {};

    issue_chunk(0, 0);

    for (int c = 0; c < NCHUNK; ++c) {
        if (c + 1 < NCHUNK) {
            issue_chunk((c + 1) * KC, (c + 1) & 1);
            wait_async_le8();             // previous batch (current buffer) done
        } else {
            wait_async_0();
        }
        __syncthreads();                  // all waves' copies visible

        const char* xcur = (c & 1) ? xbuf[1] : xbuf[0];
        const char* wcur = (c & 1) ? wbuf[1] : wbuf[0];
        const unsigned ha = (unsigned)(half * 8);

        #pragma unroll
        for (int k = 0; k < KC; k += 4) {
            // A fragment (16x4 f32): lanes 0-15 -> (K,K+1), 16-31 -> (K+2,K+3)
            v2f a = *(const v2f*)(xcur + (wave * 16 + l16) * XROW_B
                                       + (k >> 2) * 16 + ha);
            // B fragments (4x16 f32), mirrored K striping across lane halves
            const char* wk = wcur + (k >> 2) * WROW_B + ha;
            v2f b0 = *(const v2f*)(wk + ( 0 + l16) * 16);
            v2f b1 = *(const v2f*)(wk + (16 + l16) * 16);
            v2f b2 = *(const v2f*)(wk + (32 + l16) * 16);
            v2f b3 = *(const v2f*)(wk + (48 + l16) * 16);
            acc0 = __builtin_amdgcn_wmma_f32_16x16x4_f32(false, a, false, b0,
                                                         (short)0, acc0, false, false);
            acc1 = __builtin_amdgcn_wmma_f32_16x16x4_f32(false, a, false, b1,
                                                         (short)0, acc1, false, false);
            acc2 = __builtin_amdgcn_wmma_f32_16x16x4_f32(false, a, false, b2,
                                                         (short)0, acc2, false, false);
            acc3 = __builtin_amdgcn_wmma_f32_16x16x4_f32(false, a, false, b3,
                                                         (short)0, acc3, false, false);
        }
        __syncthreads();                  // reads done before buffer reuse
    }

    // ---- scatter C (VGPR r, lane l -> M = r + 8*half, N = t*16 + l16) ----
    #pragma unroll
    for (int r = 0; r < 8; ++r) {
        int m = wave * 16 + r + 8 * half;
        scores[m][ 0 + l16] = acc0[r];
        scores[m][16 + l16] = acc1[r];
        scores[m][32 + l16] = acc2[r];
        scores[m][48 + l16] = acc3[r];
    }
    __syncthreads();

    // ---- per-row top-8 (sorted desc, first-index tie-break) + renorm softmax
    if (tid < ROWS) {
        const float* sv = scores[tid];
        unsigned long long used = 0ull;
        float vals[TOPK];
        int   inds[TOPK];
        #pragma unroll
        for (int s = 0; s < TOPK; ++s) {
            float best = -3.402823466e38f;
            int   bi   = 0;
            for (int e = 0; e < E_EXP; ++e) {
                float v  = sv[e];
                bool  ok = (((used >> e) & 1ull) == 0ull) && (v > best);
                best = ok ? v : best;
                bi   = ok ? e : bi;
            }
            used |= (1ull << bi);
            vals[s] = best;
            inds[s] = bi;
        }
        const float mx = vals[0];
        float ex[TOPK];
        float sum = 0.0f;
        #pragma unroll
        for (int s = 0; s < TOPK; ++s) {
            ex[s] = __expf(vals[s] - mx);
            sum  += ex[s];
        }
        const float inv = 1.0f / sum;
        const int row = row0 + tid;
        #pragma unroll
        for (int s = 0; s < TOPK; ++s) {
            idx_out[row * TOPK + s] = inds[s];
            w_out  [row * TOPK + s] = ex[s] * inv;
        }
    }
}

extern "C" void kernel_launch(void* const* d_in, const int* in_sizes, int n_in,
                              void* d_out, int out_size, void* d_ws, size_t ws_size,
                              hipStream_t stream) {
    const float* x = (const float*)d_in[0];   // hidden_states (4,4096,2048) f32
    const float* w = (const float*)d_in[1];   // weight (64,2048) f32

    // Tuple output, concatenated flat: [topk_idx int32 T*8][topk_weight f32 T*8]
    int*   idx_out = (int*)d_out;
    float* w_out   = (float*)d_out + (size_t)T_TOKENS * TOPK;

    dim3 grid(T_TOKENS / ROWS);   // 256 blocks
    dim3 block(128);              // 4 wave32 waves
    moe_gate_topk_kernel<<<grid, block, 0, stream>>>(x, w, idx_out, w_out);
}